// QM9_Net_89515708383770
// MI455X (gfx1250) — compile-verified
//
#include <hip/hip_runtime.h>
#include <hip/hip_bf16.h>
#include <math.h>

// ---------------------------------------------------------------------------
// QM9-style GIN network for gfx1250 (MI455X).
//  - edge message/scatter: fused bond-linear + gelu + weighted scatter-add
//    (avoids materializing the 410MB edge_emb tensor)
//  - node GEMMs (Nx128 @ 128x128): V_WMMA_F32_16X16X4_F32, fp32-exact
//  - BN: two-pass column stats with atomics, apply fused with ReLU (+pool)
// Input order (setup_inputs dict, params dict flattened in insertion order):
//  0 x(N,16) 1 edge_index(2,E) 2 edge_attr(E,4) 3 edge_weight(E) 4 batch(N)
//  5 enc_W 6 enc_b 7 init_W 8 init_b 9 bond_W 10 bond_b 11 eps
//  12 W1 13 b1 14 g1 15 be1 16 W2 17 b2 18 g2 19 be2 20 lin_W 21 lin_b
// ---------------------------------------------------------------------------

#define D_DIM 128
#define FN_DIM 16
#define FE_DIM 4
#define C_DIM 12
#define L_DIM 4
#define BN_EPS 1e-5f

typedef float v2f __attribute__((ext_vector_type(2)));
typedef float v8f __attribute__((ext_vector_type(8)));

// ------------------------------ utility ------------------------------------

__global__ void zero_kernel(float* __restrict__ p, long long n) {
  long long i = (long long)blockIdx.x * blockDim.x + threadIdx.x;
  long long stride = (long long)gridDim.x * blockDim.x;
  for (; i < n; i += stride) p[i] = 0.0f;
}

// ------------------------------ encoder ------------------------------------
// h[n][d] = enc_b[d] + sum_k x[n][k]*enc_W[k][d]
__global__ void encoder_kernel(const float* __restrict__ x,
                               const float* __restrict__ W,
                               const float* __restrict__ b,
                               float* __restrict__ h, int n) {
  long long idx = (long long)blockIdx.x * blockDim.x + threadIdx.x;
  if (idx >= (long long)n * D_DIM) return;
  int d = (int)(idx & (D_DIM - 1));
  int row = (int)(idx >> 7);
  float acc = b[d];
  const float* xr = x + (size_t)row * FN_DIM;
#pragma unroll
  for (int k = 0; k < FN_DIM; ++k) acc += xr[k] * W[k * D_DIM + d];
  h[idx] = acc;
}

// ----------------------- init readout (h @ init_W) -------------------------
// atomicAdd(out[batch[n]][c], init_b[c] + sum_k h[n][k]*init_W[k][c])
__global__ void init_out_kernel(const float* __restrict__ h,
                                const float* __restrict__ W,
                                const float* __restrict__ b,
                                const int* __restrict__ batch,
                                float* __restrict__ out, int n) {
  long long idx = (long long)blockIdx.x * blockDim.x + threadIdx.x;
  if (idx >= (long long)n * C_DIM) return;
  int c = (int)(idx % C_DIM);
  int row = (int)(idx / C_DIM);
  float acc = b[c];
  const float* hr = h + (size_t)row * D_DIM;
#pragma unroll 8
  for (int k = 0; k < D_DIM; ++k) acc += hr[k] * W[k * C_DIM + c];
  atomicAdd(out + (size_t)batch[row] * C_DIM + c, acc);
}

// ------------------------- edge message + scatter ---------------------------
// One wave (32 lanes) per edge; lane handles 4 consecutive features.
// m = gelu_exact(h[src] + (edge_attr @ bond_W + bond_b)) * ew  -> atomic agg[dst]
__global__ void message_kernel(const float* __restrict__ h,
                               const int* __restrict__ src,
                               const int* __restrict__ dst,
                               const float* __restrict__ eattr,
                               const float* __restrict__ bondW,
                               const float* __restrict__ bondb,
                               const float* __restrict__ ew,
                               float* __restrict__ agg, int ecount) {
  long long gid = (long long)blockIdx.x * blockDim.x + threadIdx.x;
  int e = (int)(gid >> 5);
  if (e >= ecount) return;
  int lane = (int)(gid & 31);
  int d0 = lane * 4;
  int s = src[e];
  int d = dst[e];
  float w = ew[e];
  float a0 = eattr[(size_t)e * 4 + 0];
  float a1 = eattr[(size_t)e * 4 + 1];
  float a2 = eattr[(size_t)e * 4 + 2];
  float a3 = eattr[(size_t)e * 4 + 3];
  const float4 hv4 = *reinterpret_cast<const float4*>(h + (size_t)s * D_DIM + d0);
  float hv[4] = {hv4.x, hv4.y, hv4.z, hv4.w};
#pragma unroll
  for (int i = 0; i < 4; ++i) {
    int dd = d0 + i;
    float eb = bondb[dd] + a0 * bondW[0 * D_DIM + dd] + a1 * bondW[1 * D_DIM + dd] +
               a2 * bondW[2 * D_DIM + dd] + a3 * bondW[3 * D_DIM + dd];
    float xv = hv[i] + eb;
    float g = 0.5f * xv * (1.0f + erff(xv * 0.7071067811865476f));
    atomicAdd(agg + (size_t)d * D_DIM + dd, g * w);
  }
}

// --------------------------- WMMA fp32 GEMM --------------------------------
// out[n][:] = A[n][:] @ W + bias, A is (1+eps)*h + agg (mode 1) or src (mode 0).
// Block: 256 threads = 8 waves. Block owns 16 rows; wave w owns cols [16w,16w+16).
// V_WMMA_F32_16X16X4_F32 fragment layout (ISA 7.12.2):
//   A 16x4:  lanes 0-15 -> M=lane, K={0,1}; lanes 16-31 -> M=lane-16, K={2,3}
//   B 4x16:  v0 = B[K0][n], v1 = B[K0+1][n], n=lane&15, K0 = lane<16 ? 0 : 2
//   D 16x16: vgpr r -> M = r (lanes 0-15) / r+8 (lanes 16-31), N = lane&15
__global__ __launch_bounds__(256) void gemm_wmma_kernel(
    const float* __restrict__ A0, const float* __restrict__ A1,
    const float* __restrict__ epsArr, int layer, int mode,
    const float* __restrict__ W, const float* __restrict__ bias,
    float* __restrict__ out, int nrows) {
  __shared__ float sA[16 * D_DIM];  // 8 KB staged A tile
  const int tid = threadIdx.x;
  const int rowBase = blockIdx.x * 16;
  const float epsv = mode ? (1.0f + epsArr[layer]) : 0.0f;

  // cooperative A tile load (fused (1+eps)*h + agg for mode 1)
  for (int i = tid; i < 16 * D_DIM; i += 256) {
    int r = i >> 7;
    int c = i & (D_DIM - 1);
    int row = rowBase + r;
    float v = 0.0f;
    if (row < nrows) {
      size_t gi = (size_t)row * D_DIM + c;
      v = A0[gi];
      if (mode) v = epsv * v + A1[gi];
    }
    sA[i] = v;
  }
  __syncthreads();

  const int wave = tid >> 5;
  const int lane = tid & 31;
  const int colBase = wave * 16;
  const int nIdx = lane & 15;
  const int kHalf = (lane < 16) ? 0 : 2;

  v8f acc = {};
#pragma unroll
  for (int kk = 0; kk < 32; ++kk) {
    const int k = kk * 4;
    v2f a, b;
    a.x = sA[nIdx * D_DIM + k + kHalf];
    a.y = sA[nIdx * D_DIM + k + kHalf + 1];
    b.x = W[(size_t)(k + kHalf) * D_DIM + colBase + nIdx];
    b.y = W[(size_t)(k + kHalf + 1) * D_DIM + colBase + nIdx];
    acc = __builtin_amdgcn_wmma_f32_16x16x4_f32(
        /*neg_a=*/false, a, /*neg_b=*/false, b,
        /*c_mod=*/(short)0, acc, /*reuse_a=*/false, /*reuse_b=*/false);
  }

  const float bv = bias[colBase + nIdx];
  const int mOff = (lane < 16) ? 0 : 8;
#pragma unroll
  for (int r = 0; r < 8; ++r) {
    int row = rowBase + mOff + r;
    if (row < nrows) out[(size_t)row * D_DIM + colBase + nIdx] = acc[r] + bv;
  }
}

// ----------------------------- BN statistics --------------------------------
__global__ void stats_kernel(const float* __restrict__ t, float* __restrict__ sum,
                             float* __restrict__ sq, int n) {
  const int d = threadIdx.x;  // 128 threads
  const int base = blockIdx.x * 512;
  float s = 0.0f, ss = 0.0f;
  for (int r = 0; r < 512; ++r) {
    int row = base + r;
    if (row < n) {
      float v = t[(size_t)row * D_DIM + d];
      s += v;
      ss += v * v;
    }
  }
  atomicAdd(sum + d, s);
  atomicAdd(sq + d, ss);
}

__global__ void bnparam_kernel(const float* __restrict__ sum, const float* __restrict__ sq,
                               const float* __restrict__ g, const float* __restrict__ be,
                               float* __restrict__ scale, float* __restrict__ shift,
                               float invN) {
  const int d = threadIdx.x;
  float mu = sum[d] * invN;
  float var = sq[d] * invN - mu * mu;
  float sc = g[d] * rsqrtf(var + BN_EPS);
  scale[d] = sc;
  shift[d] = be[d] - mu * sc;
}

// relu(bn(t)) -> outh; optionally pool into pooled[batch[row]]
__global__ void bnapply_kernel(const float* __restrict__ t, const float* __restrict__ scale,
                               const float* __restrict__ shift, float* __restrict__ outh,
                               float* __restrict__ pooled, const int* __restrict__ batch,
                               int n, int doPool) {
  long long idx = (long long)blockIdx.x * blockDim.x + threadIdx.x;
  if (idx >= (long long)n * D_DIM) return;
  int d = (int)(idx & (D_DIM - 1));
  int row = (int)(idx >> 7);
  float v = t[idx] * scale[d] + shift[d];
  v = fmaxf(v, 0.0f);
  outh[idx] = v;
  if (doPool) atomicAdd(pooled + (size_t)batch[row] * D_DIM + d, v);
}

// ----------------------- layer readout (pooled @ lin_W) --------------------
__global__ void outproj_kernel(const float* __restrict__ pooled,
                               const float* __restrict__ linW,
                               const float* __restrict__ linb,
                               float* __restrict__ out, int gcount) {
  long long idx = (long long)blockIdx.x * blockDim.x + threadIdx.x;
  if (idx >= (long long)gcount * C_DIM) return;
  int c = (int)(idx % C_DIM);
  int g = (int)(idx / C_DIM);
  float acc = linb[c];
  const float* pr = pooled + (size_t)g * D_DIM;
#pragma unroll 8
  for (int k = 0; k < D_DIM; ++k) acc += pr[k] * linW[k * C_DIM + c];
  out[idx] += acc;
}

// ------------------------------ launcher -----------------------------------

extern "C" void kernel_launch(void* const* d_in, const int* in_sizes, int n_in,
                              void* d_out, int out_size, void* d_ws, size_t ws_size,
                              hipStream_t stream) {
  const float* x     = (const float*)d_in[0];
  const int*   eidx  = (const int*)d_in[1];
  const float* eattr = (const float*)d_in[2];
  const float* ew    = (const float*)d_in[3];
  const int*   batch = (const int*)d_in[4];
  const float* encW  = (const float*)d_in[5];
  const float* encb  = (const float*)d_in[6];
  const float* initW = (const float*)d_in[7];
  const float* initb = (const float*)d_in[8];
  const float* bondW = (const float*)d_in[9];
  const float* bondb = (const float*)d_in[10];
  const float* eps   = (const float*)d_in[11];
  const float* W1    = (const float*)d_in[12];
  const float* b1    = (const float*)d_in[13];
  const float* g1    = (const float*)d_in[14];
  const float* be1   = (const float*)d_in[15];
  const float* W2    = (const float*)d_in[16];
  const float* b2    = (const float*)d_in[17];
  const float* g2    = (const float*)d_in[18];
  const float* be2   = (const float*)d_in[19];
  const float* linW  = (const float*)d_in[20];
  const float* linb  = (const float*)d_in[21];

  const int N = in_sizes[0] / FN_DIM;   // 50000
  const int E = in_sizes[3];            // 800000
  const int G = out_size / C_DIM;       // 2500

  const long long ND = (long long)N * D_DIM;
  const long long GD = (long long)G * D_DIM;

  float* ws     = (float*)d_ws;
  float* h      = ws;
  float* agg    = h + ND;
  float* t1     = agg + ND;
  float* t2     = t1 + ND;
  float* pooled = t2 + ND;
  float* sum    = pooled + GD;
  float* sq     = sum + D_DIM;
  float* scale  = sq + D_DIM;
  float* shift  = scale + D_DIM;
  float* out    = (float*)d_out;

  const int*   src = eidx;
  const int*   dst = eidx + E;

  const int TB = 256;
  const int rowTiles   = (N + 15) / 16;
  const int statBlocks = (N + 511) / 512;
  const float invN = 1.0f / (float)N;

  auto grid1 = [&](long long total) { return (int)((total + TB - 1) / TB); };

  // out = 0
  zero_kernel<<<grid1((long long)G * C_DIM), TB, 0, stream>>>(out, (long long)G * C_DIM);
  // h = x @ enc_W + enc_b
  encoder_kernel<<<grid1(ND), TB, 0, stream>>>(x, encW, encb, h, N);
  // out += segsum(h @ init_W + init_b)
  init_out_kernel<<<grid1((long long)N * C_DIM), TB, 0, stream>>>(h, initW, initb, batch, out, N);

  for (int l = 0; l < L_DIM; ++l) {
    // agg = 0 ; pooled = 0 ; stats = 0
    zero_kernel<<<grid1(ND), TB, 0, stream>>>(agg, ND);
    zero_kernel<<<grid1(GD), TB, 0, stream>>>(pooled, GD);
    zero_kernel<<<1, 256, 0, stream>>>(sum, 2 * D_DIM);

    // message + scatter-add (bond-linear fused, gelu exact, edge-weighted)
    message_kernel<<<grid1((long long)E * 32), TB, 0, stream>>>(
        h, src, dst, eattr, bondW, bondb, ew, agg, E);

    // t1 = ((1+eps_l)*h + agg) @ W1[l] + b1[l]   (WMMA fp32)
    gemm_wmma_kernel<<<rowTiles, TB, 0, stream>>>(
        h, agg, eps, l, /*mode=*/1, W1 + (size_t)l * D_DIM * D_DIM,
        b1 + (size_t)l * D_DIM, t1, N);

    // BN1 + ReLU (in place)
    stats_kernel<<<statBlocks, D_DIM, 0, stream>>>(t1, sum, sq, N);
    bnparam_kernel<<<1, D_DIM, 0, stream>>>(sum, sq, g1 + l * D_DIM, be1 + l * D_DIM,
                                            scale, shift, invN);
    bnapply_kernel<<<grid1(ND), TB, 0, stream>>>(t1, scale, shift, t1, pooled, batch, N, 0);

    // t2 = t1 @ W2[l] + b2[l]   (WMMA fp32)
    gemm_wmma_kernel<<<rowTiles, TB, 0, stream>>>(
        t1, nullptr, eps, l, /*mode=*/0, W2 + (size_t)l * D_DIM * D_DIM,
        b2 + (size_t)l * D_DIM, t2, N);

    // BN2 + ReLU -> h, fused pooling
    zero_kernel<<<1, 256, 0, stream>>>(sum, 2 * D_DIM);
    stats_kernel<<<statBlocks, D_DIM, 0, stream>>>(t2, sum, sq, N);
    bnparam_kernel<<<1, D_DIM, 0, stream>>>(sum, sq, g2 + l * D_DIM, be2 + l * D_DIM,
                                            scale, shift, invN);
    bnapply_kernel<<<grid1(ND), TB, 0, stream>>>(t2, scale, shift, h, pooled, batch, N, 1);

    // out += pooled @ lin_W[l] + lin_b[l]
    outproj_kernel<<<grid1((long long)G * C_DIM), TB, 0, stream>>>(
        pooled, linW + (size_t)l * D_DIM * C_DIM, linb + (size_t)l * C_DIM, out, G);
  }
  (void)n_in; (void)ws_size;
}